// RNN_57647051047009
// MI455X (gfx1250) — compile-verified
//
#include <hip/hip_runtime.h>
#include <hip/hip_bf16.h>

typedef __attribute__((ext_vector_type(2))) float v2f;
typedef __attribute__((ext_vector_type(8))) float v8f;

#define F_IN   256
#define H_DIM  256
#define F_OUT  128
#define T_STEPS 131072

// ---------------------------------------------------------------------------
// Kernel A: xw[t][j] = sum_k x[t][k] * W_x[j][k] + b_hid[j]
// using V_WMMA_F32_16X16X4_F32 (full fp32 matrix path).
// grid = (T/16, 2), block = 256 threads (8 waves). Wave w handles the 16x16
// output tile (t0 = bx*16, j0 = (by*8+w)*16), accumulating over K=256 in
// 64 WMMA steps of K=4.
// ---------------------------------------------------------------------------
__global__ __launch_bounds__(256)
void rnn_xw_wmma(const float* __restrict__ x,
                 const float* __restrict__ W_hid,
                 const float* __restrict__ b_hid,
                 float* __restrict__ xw)
{
    const int lane = threadIdx.x & 31;
    const int wave = threadIdx.x >> 5;
    const int t0 = blockIdx.x * 16;
    const int j0 = (blockIdx.y * 8 + wave) * 16;

    const int m  = lane & 15;          // A row (t) / B col (j) for this lane
    const int kk = (lane >> 4) * 2;    // K sub-offset: lanes 0-15 -> K 0,1 ; 16-31 -> K 2,3

    // A: x[t0+m][k0+kk .. +1]   (row-major, stride F_IN)
    const float* xrow = x + (size_t)(t0 + m) * F_IN + kk;
    // B: W_x[j0+m][k0+kk .. +1] = W_hid[j0+m][k0+kk] (W_hid stride F_IN+H, first 256 cols)
    const float* wrow = W_hid + (size_t)(j0 + m) * (F_IN + H_DIM) + kk;

    // C/D layout: VGPR r, lane L -> (M = r + 8*(L>>4), N = L&15). All 8 share N,
    // so the bias can seed the accumulator directly.
    const float bias = b_hid[j0 + m];
    v8f c;
#pragma unroll
    for (int r = 0; r < 8; ++r) c[r] = bias;

#pragma unroll 8
    for (int k0 = 0; k0 < F_IN; k0 += 4) {
        v2f a = *(const v2f*)(xrow + k0);   // 8B aligned (kk even, k0 % 4 == 0)
        v2f b = *(const v2f*)(wrow + k0);
        c = __builtin_amdgcn_wmma_f32_16x16x4_f32(
                /*neg_a=*/false, a, /*neg_b=*/false, b,
                /*c_mod=*/(short)0, c, /*reuse_a=*/false, /*reuse_b=*/false);
    }

    const int nn    = lane & 15;
    const int mbase = t0 + 8 * (lane >> 4);
#pragma unroll
    for (int r = 0; r < 8; ++r) {
        xw[(size_t)(mbase + r) * H_DIM + (j0 + nn)] = c[r];
    }
}

// ---------------------------------------------------------------------------
// Kernel B: the sequential recurrence, single workgroup (256 threads = 8 waves)
// on one WGP. Thread j keeps W_h row j (256 f32) entirely in VGPRs; h is
// double-buffered in LDS so exactly one barrier per step. xw_t for the next
// step is prefetched into a register to hide HBM latency behind the 256-FMA
// dot product. Final projection (h_final @ W_out.T + b_out) folded in.
// ---------------------------------------------------------------------------
__global__ __launch_bounds__(256)
void rnn_recurrence(const float* __restrict__ xw,
                    const float* __restrict__ W_hid,
                    const float* __restrict__ W_out,
                    const float* __restrict__ b_out,
                    float* __restrict__ out)
{
    __shared__ float h_s[2][H_DIM];

    const int j = threadIdx.x;   // 0..255 : output index of the matvec

    // --- stage W_h row j into registers (one-time) ---
    float w[H_DIM];
    {
        const float* wr = W_hid + (size_t)j * (F_IN + H_DIM) + F_IN;  // W_h part
#pragma unroll
        for (int k = 0; k < H_DIM; k += 4) {
            float4 t = *(const float4*)(wr + k);
            w[k + 0] = t.x; w[k + 1] = t.y; w[k + 2] = t.z; w[k + 3] = t.w;
        }
    }

    h_s[0][j] = 0.0f;            // h0 = 0
    float xwv = xw[j];           // xw_0[j] preloaded
    __syncthreads();

    for (int t = 0; t < T_STEPS; ++t) {
        const float* h = h_s[t & 1];

        // 256-FMA dot: uniform-address ds_load_b128 broadcasts of h,
        // W operand from registers. 4 accumulator chains for ILP.
        float a0 = 0.f, a1 = 0.f, a2 = 0.f, a3 = 0.f;
#pragma unroll
        for (int k = 0; k < H_DIM; k += 4) {
            float4 hv = *(const float4*)(h + k);
            a0 = fmaf(hv.x, w[k + 0], a0);
            a1 = fmaf(hv.y, w[k + 1], a1);
            a2 = fmaf(hv.z, w[k + 2], a2);
            a3 = fmaf(hv.w, w[k + 3], a3);
        }

        float hn = tanhf(xwv + ((a0 + a1) + (a2 + a3)));
        h_s[(t & 1) ^ 1][j] = hn;

        // prefetch next timestep's xw while other waves finish
        if (t + 1 < T_STEPS) xwv = xw[(size_t)(t + 1) * H_DIM + j];

        __syncthreads();
    }

    // --- final projection: out[o] = b_out[o] + sum_k h_final[k] * W_out[o][k] ---
    if (j < F_OUT) {
        const float* hf = h_s[T_STEPS & 1];   // T even -> buffer 0
        const float* wo = W_out + (size_t)j * H_DIM;
        float acc = b_out[j];
#pragma unroll 8
        for (int k = 0; k < H_DIM; ++k) acc = fmaf(hf[k], wo[k], acc);
        out[j] = acc;
    }
}

// ---------------------------------------------------------------------------
// Launch: inputs per setup_inputs() order: x, W_hid, b_hid, W_out, b_out.
// d_ws holds xw (T*H f32 = 128 MiB).
// ---------------------------------------------------------------------------
extern "C" void kernel_launch(void* const* d_in, const int* in_sizes, int n_in,
                              void* d_out, int out_size, void* d_ws, size_t ws_size,
                              hipStream_t stream)
{
    const float* x     = (const float*)d_in[0];
    const float* W_hid = (const float*)d_in[1];
    const float* b_hid = (const float*)d_in[2];
    const float* W_out = (const float*)d_in[3];
    const float* b_out = (const float*)d_in[4];
    float* out = (float*)d_out;
    float* xw  = (float*)d_ws;

    dim3 grid(T_STEPS / 16, 2);
    rnn_xw_wmma<<<grid, 256, 0, stream>>>(x, W_hid, b_hid, xw);
    rnn_recurrence<<<1, 256, 0, stream>>>(xw, W_hid, W_out, b_out, out);
}